// TropicalLinear_7524782702954
// MI455X (gfx1250) — compile-verified
//
#include <hip/hip_runtime.h>
#include <stdint.h>

// Tropical (max-plus) linear map: y[b,o] = max_i (x[b,i] + W[o,i])
// B=512, I=1024, O=1024, fp32. VALU-bound => TDM feeds LDS, VALU does max-plus.
// KT=64 halves barrier/tensorcnt overhead vs KT=32; TDM pad keeps LDS banks clean.

#define B_SZ 512
#define I_SZ 1024
#define O_SZ 1024
#define KT   64            // K-chunk (dwords per tile row)
#define ROWS 64            // tile rows (b-rows for xs, o-rows for ws)
#define PAD  4             // pad dwords per row, inserted by TDM pad_enable
#define RS   (KT + PAD)    // LDS row stride in dwords (68 -> 272B, 16B aligned)

typedef unsigned int u32x4 __attribute__((ext_vector_type(4)));
typedef int          i32x8 __attribute__((ext_vector_type(8)));
typedef int          i32x4 __attribute__((ext_vector_type(4)));
typedef float        f32x4 __attribute__((ext_vector_type(4)));

// Issue one TDM load of a ROWS x KT fp32 tile (row stride I_SZ elements in
// global) into LDS at lds_byte_off, padding each 64-dword row by 4 dwords.
__device__ __forceinline__ void tdm_load_tile(unsigned lds_byte_off,
                                              const float* gptr,
                                              unsigned tensor_rows)
{
  uint64_t ga = (uint64_t)(uintptr_t)gptr;

  u32x4 g0;
  g0[0] = 1u;                                               // count=1 (valid user D#)
  g0[1] = lds_byte_off;                                     // lds_addr [63:32]
  g0[2] = (unsigned)(ga & 0xFFFFFFFFu);                     // global_addr[31:0]
  g0[3] = (unsigned)((ga >> 32) & 0x01FFFFFFu) | 0x80000000u; // addr[56:32] | type=2

  i32x8 g1;
  // data_size=2 (4B), pad_enable=1, pad_interval=5 (64 dwords), pad_amount=3 (4 dwords)
  g1[0] = (int)((2u << 16) | (1u << 20) | (5u << 22) | (3u << 25));
  g1[1] = (int)(((unsigned)I_SZ & 0xFFFFu) << 16);          // tensor_dim0[15:0] @ bits 63:48
  g1[2] = (int)((tensor_rows & 0xFFFFu) << 16);             // tensor_dim1[15:0] @ bits 95:80
  g1[3] = (int)((unsigned)KT << 16);                        // tile_dim0 @ bits 127:112
  g1[4] = (int)ROWS;                                        // tile_dim1 @ bits 143:128 (tile_dim2=0)
  g1[5] = (int)I_SZ;                                        // tensor_dim0_stride[31:0]
  g1[6] = 0;                                                // stride hi / dim1_stride lo
  g1[7] = 0;

  i32x4 z4 = {0, 0, 0, 0};
#if defined(__clang_major__) && (__clang_major__ >= 23)
  i32x8 z8 = {0, 0, 0, 0, 0, 0, 0, 0};
  __builtin_amdgcn_tensor_load_to_lds(g0, g1, z4, z4, z8, 0);
#else
  __builtin_amdgcn_tensor_load_to_lds(g0, g1, z4, z4, 0);
#endif
}

__global__ __launch_bounds__(256)
void TropicalLinear_kernel(const float* __restrict__ x,
                           const float* __restrict__ W,
                           float* __restrict__ out)
{
  __shared__ __align__(16) float xs[2][ROWS * RS];
  __shared__ __align__(16) float ws[2][ROWS * RS];

  const int tid = threadIdx.x;
  const int tx  = tid & 15;      // o-lane
  const int ty  = tid >> 4;      // b-group
  const int b0  = blockIdx.y * 64;
  const int o0  = blockIdx.x * 64;

  const bool issuer = (tid < 32);   // wave 0 drives the TDM

  if (issuer) {
    tdm_load_tile((unsigned)(uintptr_t)&xs[0][0], x + (size_t)b0 * I_SZ, B_SZ);
    tdm_load_tile((unsigned)(uintptr_t)&ws[0][0], W + (size_t)o0 * I_SZ, O_SZ);
  }

  float acc[4][4];
#pragma unroll
  for (int r = 0; r < 4; ++r)
#pragma unroll
    for (int c = 0; c < 4; ++c)
      acc[r][c] = -__builtin_inff();

  const int NC = I_SZ / KT;  // 16 chunks
  for (int kc = 0; kc < NC; ++kc) {
    const int buf = kc & 1;

    if (issuer) {
      if (kc + 1 < NC) {
        const int k0 = (kc + 1) * KT;
        tdm_load_tile((unsigned)(uintptr_t)&xs[buf ^ 1][0],
                      x + (size_t)b0 * I_SZ + k0, B_SZ);
        tdm_load_tile((unsigned)(uintptr_t)&ws[buf ^ 1][0],
                      W + (size_t)o0 * I_SZ + k0, O_SZ);
        __builtin_amdgcn_s_wait_tensorcnt(2);   // chunk kc complete (in-order)
      } else {
        __builtin_amdgcn_s_wait_tensorcnt(0);   // last chunk complete
      }
    }
    __syncthreads();   // publish chunk kc to all waves

    const float* xrow = &xs[buf][(ty * 4) * RS];
    const float* wrow = &ws[buf][tx * RS];

#pragma unroll
    for (int kk = 0; kk < KT; kk += 4) {
      f32x4 xv[4], wv[4];
#pragma unroll
      for (int r = 0; r < 4; ++r)
        xv[r] = *(const f32x4*)&xrow[r * RS + kk];          // ds_load_b128, broadcast-ish
#pragma unroll
      for (int c = 0; c < 4; ++c)
        wv[c] = *(const f32x4*)&wrow[(c * 16) * RS + kk];   // ds_load_b128, 64-bank spread
#pragma unroll
      for (int r = 0; r < 4; ++r) {
#pragma unroll
        for (int c = 0; c < 4; ++c) {
          float m0 = __builtin_fmaxf(xv[r][0] + wv[c][0], xv[r][1] + wv[c][1]);
          float m1 = __builtin_fmaxf(xv[r][2] + wv[c][2], xv[r][3] + wv[c][3]);
          acc[r][c] = __builtin_fmaxf(acc[r][c], __builtin_fmaxf(m0, m1));
        }
      }
    }
    __syncthreads();   // all reads of buf done before wave0 refills it next iter
  }

  // Store 64x64 tile: b = b0 + ty*4 + r, o = o0 + c*16 + tx (coalesced in tx).
#pragma unroll
  for (int r = 0; r < 4; ++r) {
    const int b = b0 + ty * 4 + r;
#pragma unroll
    for (int c = 0; c < 4; ++c) {
      const int o = o0 + c * 16 + tx;
      out[(size_t)b * O_SZ + o] = acc[r][c];
    }
  }
}

extern "C" void kernel_launch(void* const* d_in, const int* in_sizes, int n_in,
                              void* d_out, int out_size, void* d_ws, size_t ws_size,
                              hipStream_t stream) {
  (void)in_sizes; (void)n_in; (void)d_ws; (void)ws_size; (void)out_size;
  const float* x = (const float*)d_in[0];   // [B, I] fp32
  const float* W = (const float*)d_in[1];   // [O, I] fp32
  float* out = (float*)d_out;               // [B, O] fp32

  dim3 grid(O_SZ / 64, B_SZ / 64, 1);       // 16 x 8 blocks
  dim3 block(256, 1, 1);                    // 8 waves
  TropicalLinear_kernel<<<grid, block, 0, stream>>>(x, W, out);
}